// GATSelfAttention_6665789243985
// MI455X (gfx1250) — compile-verified
//
#include <hip/hip_runtime.h>
#include <math.h>

// Problem constants
constexpr int NB    = 32;    // batches
constexpr int EN    = 512;   // nodes
constexpr int DIM   = 300;   // in/out dim
constexpr int DIMP  = 304;   // padded to 16 (19 tiles)
constexpr int NTILES = DIMP / 16;  // 19
constexpr int NTY   = 4;     // edge types
constexpr int TWOD  = 600;   // 2*OUT_DIM
#define SLOPE_F 0.2f

typedef __attribute__((ext_vector_type(2))) float v2f;
typedef __attribute__((ext_vector_type(8))) float v8f;

// Workspace layout (float offsets)
constexpr int OFF_U1   = 0;
constexpr int OFF_U2   = OFF_U1   + NTY * NB * DIM;     //  38400
constexpr int OFF_SSRC = OFF_U2   + NTY * NB * DIM;     //  76800
constexpr int OFF_SDST = OFF_SSRC + NTY * NB * EN;      // 142336
constexpr int OFF_HM   = OFF_SDST + NTY * NB * EN;      // 207872
constexpr int OFF_COEF = OFF_HM   + NB * EN * DIMP;     // 5188608
constexpr int OFF_WPAD = OFF_COEF + NB * EN * EN;       // 13577216
// total ~13.67M floats (~54.7 MB)

// -------------------------------------------------------------------------
// Kernel 0: pad W_type[3] (300x300) into 300x304 image (zeros in pad) so the
// WMMA B-operand loads in h3_kernel are branch-free.
// -------------------------------------------------------------------------
__global__ void padW_kernel(const float* __restrict__ W_type,
                            float* __restrict__ ws) {
    const float* W3 = W_type + (NTY - 1) * DIM * DIM;
    for (int idx = blockIdx.x * 256 + threadIdx.x; idx < DIM * DIMP;
         idx += gridDim.x * 256) {
        int row = idx / DIMP, col = idx - row * DIMP;
        ws[OFF_WPAD + idx] = (col < DIM) ? W3[row * DIM + col] : 0.f;
    }
}

// -------------------------------------------------------------------------
// Kernel 1: per (type, batch): gate MLP, fold (gate * a) back through W_type
// u1[d] = sum_f W[d,f] * g1[f]*a1[f]   (so s_src = x . u1, no h needed)
// -------------------------------------------------------------------------
__global__ void gates_kernel(const float* __restrict__ query,
                             const float* __restrict__ W_type,
                             const float* __restrict__ a_type,
                             const float* __restrict__ W1,
                             const float* __restrict__ W2,
                             float* __restrict__ ws) {
    __shared__ float q[DIM];
    __shared__ float t1[TWOD];
    __shared__ float c1[DIM];
    __shared__ float c2[DIM];
    const int ty = blockIdx.x, n = blockIdx.y, tid = threadIdx.x;

    const float* qn = query + n * DIM;
    for (int d = tid; d < DIM; d += 256) q[d] = qn[d];
    __syncthreads();

    const float* w1 = W1 + ty * DIM * TWOD;
    for (int k = tid; k < TWOD; k += 256) {
        float s = 0.f;
        for (int d = 0; d < DIM; ++d) s += q[d] * w1[d * TWOD + k];
        t1[k] = s > 0.f ? s : 0.f;                     // relu
    }
    __syncthreads();

    const float* w2 = W2 + ty * TWOD * TWOD;
    const float* at = a_type + ty * TWOD;
    for (int k = tid; k < TWOD; k += 256) {
        float s = 0.f;
        for (int j = 0; j < TWOD; ++j) s += t1[j] * w2[j * TWOD + k];
        float g = 1.f / (1.f + __expf(-s));            // sigmoid
        if (k < DIM) c1[k] = g * at[k];
        else         c2[k - DIM] = g * at[k];
    }
    __syncthreads();

    const float* w  = W_type + ty * DIM * DIM;
    float* u1 = ws + OFF_U1 + (ty * NB + n) * DIM;
    float* u2 = ws + OFF_U2 + (ty * NB + n) * DIM;
    for (int d = tid; d < DIM; d += 256) {
        float s1 = 0.f, s2 = 0.f;
        const float* wr = w + d * DIM;
        for (int f = 0; f < DIM; ++f) {
            float wv = wr[f];
            s1 += wv * c1[f];
            s2 += wv * c2[f];
        }
        u1[d] = s1;
        u2[d] = s2;
    }
}

// -------------------------------------------------------------------------
// Kernel 2: s_src[n,t,e] = x[n,e,:].u1[t,n,:]   s_dst likewise (8 matvecs)
// -------------------------------------------------------------------------
__global__ void src_dst_kernel(const float* __restrict__ x,
                               float* __restrict__ ws) {
    __shared__ float u[8][DIM];
    const int n = blockIdx.y, tid = threadIdx.x;
    for (int idx = tid; idx < 8 * DIM; idx += 256) {
        int t = idx / DIM, d = idx - t * DIM;
        int ty = t >> 1;
        const float* base = ws + ((t & 1) ? OFF_U2 : OFF_U1) + (ty * NB + n) * DIM;
        u[t][d] = base[d];
    }
    __syncthreads();

    const int e = blockIdx.x * 256 + tid;
    const float* xr = x + (n * EN + e) * DIM;
    float acc[8] = {0.f, 0.f, 0.f, 0.f, 0.f, 0.f, 0.f, 0.f};
    for (int d = 0; d < DIM; ++d) {
        float xv = xr[d];
#pragma unroll
        for (int t = 0; t < 8; ++t) acc[t] += xv * u[t][d];
    }
#pragma unroll
    for (int ty = 0; ty < NTY; ++ty) {
        ws[OFF_SSRC + (n * NTY + ty) * EN + e] = acc[2 * ty];
        ws[OFF_SDST + (n * NTY + ty) * EN + e] = acc[2 * ty + 1];
    }
}

// -------------------------------------------------------------------------
// Kernel 3: hm[n,e,f] = (x[n] @ W_type[3])[e,f] * mask[n,e]  via WMMA f32.
// A slab (16 rows x 300) staged once in LDS, shared by all 8 waves; B read
// from the pre-padded 300x304 W image (branch-free global loads).
// -------------------------------------------------------------------------
__global__ void h3_kernel(const float* __restrict__ x,
                          const float* __restrict__ node_mask,
                          float* __restrict__ ws) {
    __shared__ float As[16 * DIM];                     // 19.2 KB
    const int lane  = threadIdx.x & 31;
    const int wave  = threadIdx.x >> 5;
    const int ntile = blockIdx.y * 8 + wave;
    const int n      = blockIdx.z;
    const int m_base = blockIdx.x * 16;

    // cooperative stage of the A slab (coalesced)
    const float* A = x + (n * EN + m_base) * DIM;      // 16 x 300
    for (int idx = threadIdx.x; idx < 16 * DIM; idx += 256) As[idx] = A[idx];
    __syncthreads();

    if (ntile < NTILES) {                              // wave-uniform: EXEC full
        const float* Bp = ws + OFF_WPAD;               // 300 x 304 padded
        const int arow = (lane & 15) * DIM;
        const int ncol = ntile * 16 + (lane & 15);
        const int koff = (lane >> 4) << 1;             // lanes 16-31 hold K=2,3

        v8f acc = {};
        for (int k = 0; k < DIM; k += 4) {
            int kr = k + koff;
            v2f a;
            a.x = As[arow + kr];
            a.y = As[arow + kr + 1];
            v2f b;
            b.x = Bp[kr * DIMP + ncol];
            b.y = Bp[(kr + 1) * DIMP + ncol];
            // stream-ahead hint on B (global_prefetch_b8), kept in-bounds
            __builtin_prefetch(&Bp[(kr + 16 < DIM ? kr + 16 : kr) * DIMP + ncol], 0, 0);
            acc = __builtin_amdgcn_wmma_f32_16x16x4_f32(false, a, false, b,
                                                        (short)0, acc, false, false);
        }

        const int rbase = m_base + ((lane >> 4) << 3); // upper half-wave: M += 8
        float* hm = ws + OFF_HM + n * EN * DIMP;
#pragma unroll
        for (int v = 0; v < 8; ++v) {
            int row = rbase + v;
            float m = node_mask[n * EN + row];
            hm[row * DIMP + ntile * 16 + (lane & 15)] = acc[v] * m;
        }
    }
}

// -------------------------------------------------------------------------
// Kernel 4: one block per output row (n,i): adj-gathered leaky-relu scores,
// row softmax -> coefs workspace. adj==0 rows fall out as uniform (matches jax).
// -------------------------------------------------------------------------
__global__ void coef_kernel(const int* __restrict__ adj,
                            float* __restrict__ ws) {
    __shared__ float sd[NTY][EN];     // 8 KB
    __shared__ float sc[EN];          // 2 KB
    __shared__ float red[256];
    const int n = blockIdx.y, i = blockIdx.x, tid = threadIdx.x;

    for (int idx = tid; idx < NTY * EN; idx += 256) {
        int t = idx >> 9, j = idx & (EN - 1);
        sd[t][j] = ws[OFF_SDST + (n * NTY + t) * EN + j];
    }
    float ssv[NTY];
#pragma unroll
    for (int t = 0; t < NTY; ++t) ssv[t] = ws[OFF_SSRC + (n * NTY + t) * EN + i];
    __syncthreads();

    const int* arow = adj + (n * EN + i) * EN;
    float lmax = -3.0e38f;
    for (int j = tid; j < EN; j += 256) {
        int t = arow[j];
        float s;
        if (t > 0) {
            float v = ssv[t - 1] + sd[t - 1][j];
            s = v >= 0.f ? v : SLOPE_F * v;            // leaky relu
        } else {
            s = -1e30f;
        }
        sc[j] = s;
        lmax = fmaxf(lmax, s);
    }
    red[tid] = lmax;
    __syncthreads();
    for (int off = 128; off > 0; off >>= 1) {
        if (tid < off) red[tid] = fmaxf(red[tid], red[tid + off]);
        __syncthreads();
    }
    const float mx = red[0];
    __syncthreads();

    float lsum = 0.f;
    for (int j = tid; j < EN; j += 256) {
        float e = __expf(sc[j] - mx);
        sc[j] = e;
        lsum += e;
    }
    red[tid] = lsum;
    __syncthreads();
    for (int off = 128; off > 0; off >>= 1) {
        if (tid < off) red[tid] += red[tid + off];
        __syncthreads();
    }
    const float inv = 1.f / red[0];

    float* crow = ws + OFF_COEF + (n * EN + i) * EN;
    for (int j = tid; j < EN; j += 256) crow[j] = sc[j] * inv;
}

// -------------------------------------------------------------------------
// Kernel 5: out = coefs @ hm  via WMMA f32 16x16x4. K=512 -> 128 WMMAs/tile.
// A slab (16 x 512 coefs) staged once in LDS (32 KB), shared by all 8 waves.
// -------------------------------------------------------------------------
__global__ void out_gemm_kernel(const float* __restrict__ ws,
                                float* __restrict__ out) {
    __shared__ float As[16 * EN];                      // 32 KB
    const int lane  = threadIdx.x & 31;
    const int wave  = threadIdx.x >> 5;
    const int ntile = blockIdx.y * 8 + wave;
    const int n      = blockIdx.z;
    const int m_base = blockIdx.x * 16;

    const float* A = ws + OFF_COEF + (n * EN + m_base) * EN;  // 16 x 512
    for (int idx = threadIdx.x; idx < 16 * EN; idx += 256) As[idx] = A[idx];
    __syncthreads();

    if (ntile < NTILES) {                              // wave-uniform
        const float* B = ws + OFF_HM + n * EN * DIMP;  // 512 x 304 (padded)
        const int arow = (lane & 15) * EN;
        const int ncol = ntile * 16 + (lane & 15);
        const int koff = (lane >> 4) << 1;

        v8f acc = {};
        for (int k = 0; k < EN; k += 4) {
            int kr = k + koff;
            v2f a;
            a.x = As[arow + kr];
            a.y = As[arow + kr + 1];
            v2f b;
            b.x = B[kr * DIMP + ncol];
            b.y = B[(kr + 1) * DIMP + ncol];
            __builtin_prefetch(&B[(kr + 16 < EN ? kr + 16 : kr) * DIMP + ncol], 0, 0);
            acc = __builtin_amdgcn_wmma_f32_16x16x4_f32(false, a, false, b,
                                                        (short)0, acc, false, false);
        }

        if (ncol < DIM) {
            const int rbase = m_base + ((lane >> 4) << 3);
#pragma unroll
            for (int v = 0; v < 8; ++v) {
                out[(n * EN + rbase + v) * DIM + ncol] = acc[v];
            }
        }
    }
}

// -------------------------------------------------------------------------
extern "C" void kernel_launch(void* const* d_in, const int* in_sizes, int n_in,
                              void* d_out, int out_size, void* d_ws, size_t ws_size,
                              hipStream_t stream) {
    const float* input_state = (const float*)d_in[0];
    const int*   adj         = (const int*)  d_in[1];
    const float* node_mask   = (const float*)d_in[2];
    const float* query_vec   = (const float*)d_in[3];
    const float* W_type      = (const float*)d_in[4];
    const float* a_type      = (const float*)d_in[5];
    const float* qattn_W1    = (const float*)d_in[6];
    const float* qattn_W2    = (const float*)d_in[7];
    float* out = (float*)d_out;
    float* ws  = (float*)d_ws;

    (void)in_sizes; (void)n_in; (void)out_size; (void)ws_size;

    // 0) pad W_type[3] into 300x304 image (zeros in pad)
    padW_kernel<<<dim3(64), 256, 0, stream>>>(W_type, ws);
    // 1) gate MLP + fold into u-vectors       (4 x 32 blocks)
    gates_kernel<<<dim3(NTY, NB), 256, 0, stream>>>(query_vec, W_type, a_type,
                                                    qattn_W1, qattn_W2, ws);
    // 2) s_src / s_dst matvecs                (2 x 32 blocks)
    src_dst_kernel<<<dim3(EN / 256, NB), 256, 0, stream>>>(input_state, ws);
    // 3) hm = (x @ W3) * mask  [WMMA]         (32 x 3 x 32 blocks, 8 waves)
    h3_kernel<<<dim3(EN / 16, 3, NB), 256, 0, stream>>>(input_state, node_mask, ws);
    // 4) softmax coefficients                 (512 x 32 blocks)
    coef_kernel<<<dim3(EN, NB), 256, 0, stream>>>(adj, ws);
    // 5) out = coefs @ hm  [WMMA]             (32 x 3 x 32 blocks, 8 waves)
    out_gemm_kernel<<<dim3(EN / 16, 3, NB), 256, 0, stream>>>(ws, out);
}